// Memory_9835475108444
// MI455X (gfx1250) — compile-verified
//
#include <hip/hip_runtime.h>
#include <hip/hip_bf16.h>

// Problem constants (from reference): B=256, T=16, M=256.
#define BB 256
#define TT 16
#define MM 256

typedef __attribute__((ext_vector_type(2))) float v2f;
typedef __attribute__((ext_vector_type(8))) float v8f;

__device__ __forceinline__ float relu6f(float v) {
    return fminf(fmaxf(v, 0.0f), 6.0f);
}

__device__ __forceinline__ float wave_reduce_sum(float v) {
    // wave32 butterfly reduction
    #pragma unroll
    for (int off = 16; off > 0; off >>= 1) v += __shfl_xor(v, off, 32);
    return v;
}

// -----------------------------------------------------------------------------
// Phase 1: Y0[b][t][m] = sum_j A0[b][m][j] * X[b][j][t]
//   columns t=0..15 are xs[t,b,:]; the 17th column (x_query) is folded into
//   the same pass as a VALU dot product on the A fragments already in VGPRs.
// One block per batch. 16 waves; wave w computes rows m0 = 16*w .. 16*w+15
// with two interleaved chains of V_WMMA_F32_16X16X4_F32 over K=256.
// X is staged into LDS with GLOBAL_LOAD_ASYNC_TO_LDS_B32 (ASYNCcnt path).
// -----------------------------------------------------------------------------
__global__ __launch_bounds__(512) void memory_phase1_gemm(
    const float* __restrict__ A0, const float* __restrict__ xs,
    const float* __restrict__ xq, float* __restrict__ Y0) {
    const int b   = blockIdx.x;
    const int tid = threadIdx.x;

    // X staged transposed: Xl[K][n], n=0..15 -> xs t, n=16 -> x_query.
    // Row stride 18 dwords keeps B-fragment reads bank-conflict-free.
    __shared__ float Xl[MM][18];

    // Async global->LDS scatter copy (no VGPR round-trip, tracked by ASYNCcnt).
    for (int i = tid; i < 17 * MM; i += 512) {
        const int n = i >> 8;      // vector id (t, or 16 for query)
        const int k = i & (MM - 1);
        const float* gsrc = (n < 16) ? (xs + ((size_t)n * BB + b) * MM + k)
                                     : (xq + (size_t)b * MM + k);
        // Low 32 bits of the flat shared-aperture address = LDS byte offset.
        const unsigned ldsOff = (unsigned)(uintptr_t)&Xl[k][n];
        asm volatile("global_load_async_to_lds_b32 %0, %1, off"
                     :: "v"(ldsOff), "v"((unsigned long long)(uintptr_t)gsrc)
                     : "memory");
    }
    asm volatile("s_wait_asynccnt 0x0" ::: "memory");
    __syncthreads();

    const int wave  = tid >> 5;         // 0..15  -> m tile
    const int lane  = tid & 31;
    const int m0    = wave << 4;
    const int row   = lane & 15;        // A row within tile / B column N
    const int khalf = (lane >> 4) << 1; // 0 for lanes 0-15, 2 for lanes 16-31

    const float* __restrict__ Arow =
        A0 + ((size_t)b * MM + (m0 + row)) * MM;

    // Two independent accumulation chains (even/odd K-steps) for WMMA ILP.
    v8f   acc0 = {}, acc1 = {};
    float q0 = 0.0f, q1 = 0.0f;

    for (int kb = 0; kb < MM; kb += 64) {
        // Pull the next 256B of this lane's A row toward the caches
        // (speculative; harmless past the end of the row).
        __builtin_prefetch(Arow + kb + 64, 0, 0);
        #pragma unroll
        for (int k = kb; k < kb + 64; k += 8) {
            {   // even K-step -> chain 0
                const int K0 = k + khalf;
                const v2f a = *(const v2f*)(Arow + K0);
                v2f bf;
                bf.x = Xl[K0][row];
                bf.y = Xl[K0 + 1][row];
                q0 = fmaf(a.x, Xl[K0][16], q0);
                q0 = fmaf(a.y, Xl[K0 + 1][16], q0);
                acc0 = __builtin_amdgcn_wmma_f32_16x16x4_f32(
                    false, a, false, bf, (short)0, acc0, false, false);
            }
            {   // odd K-step -> chain 1
                const int K0 = k + 4 + khalf;
                const v2f a = *(const v2f*)(Arow + K0);
                v2f bf;
                bf.x = Xl[K0][row];
                bf.y = Xl[K0 + 1][row];
                q1 = fmaf(a.x, Xl[K0][16], q1);
                q1 = fmaf(a.y, Xl[K0 + 1][16], q1);
                acc1 = __builtin_amdgcn_wmma_f32_16x16x4_f32(
                    false, a, false, bf, (short)0, acc1, false, false);
            }
        }
    }

    float* __restrict__ Yb = Y0 + (size_t)b * (17 * MM);

    // D layout: VGPR r, lanes 0-15 -> (M=r, N=lane), lanes 16-31 -> (M=r+8).
    const int half = lane >> 4;
    #pragma unroll
    for (int r = 0; r < 8; ++r) {
        const int mrow = m0 + r + 8 * half;
        Yb[row * MM + mrow] = acc0[r] + acc1[r];
    }

    // Query column: lanes r and r+16 jointly cover all K for row r.
    const float qacc = q0 + q1;
    const float qtot = qacc + __shfl_xor(qacc, 16, 32);
    if (lane < 16) {
        Yb[16 * MM + m0 + lane] = qtot;
    }
}

// -----------------------------------------------------------------------------
// Phase 2: per-batch low-rank recurrence + final readout.
//   y_t = lam^t * Y0[t] + sum_{s<t} lam^{t-1-s} * (p2_s . x_t) * x_s
//   p2_t = learn * relu6(learn2 * x_t + y_t)
//   out  = relu6(lam^16 * Y0[16] + sum_s lam^{15-s} * (p2_s . xq) * x_s)
// One wave (32 lanes) per batch; lane owns elements lane+32e, e=0..7.
// -----------------------------------------------------------------------------
__global__ __launch_bounds__(32) void memory_phase2_recur(
    const float* __restrict__ xs, const float* __restrict__ xq,
    const float* __restrict__ pdecay, const float* __restrict__ plearn,
    const float* __restrict__ plearn2, const float* __restrict__ Y0,
    float* __restrict__ out) {
    const int b    = blockIdx.x;
    const int lane = threadIdx.x;

    __shared__ float p2l[TT][MM];  // p2 history
    __shared__ float xh[TT][MM];   // x history

    const float lam    = 1.0f - pdecay[0];
    const float learn  = plearn[0];
    const float learn2 = plearn2[0];

    float lampow[TT + 1];
    lampow[0] = 1.0f;
    #pragma unroll
    for (int i = 1; i <= TT; ++i) lampow[i] = lampow[i - 1] * lam;

    const float* __restrict__ Yb = Y0 + (size_t)b * (17 * MM);

    for (int t = 0; t < TT; ++t) {
        float xt[8], y[8];
        #pragma unroll
        for (int e = 0; e < 8; ++e) {
            const int idx = lane + 32 * e;
            xt[e] = xs[((size_t)t * BB + b) * MM + idx];
            y[e]  = lampow[t] * Yb[t * MM + idx];
        }
        for (int s = 0; s < t; ++s) {
            float partial = 0.0f;
            #pragma unroll
            for (int e = 0; e < 8; ++e)
                partial += p2l[s][lane + 32 * e] * xt[e];
            const float d = wave_reduce_sum(partial);
            const float w = d * lampow[t - 1 - s];
            #pragma unroll
            for (int e = 0; e < 8; ++e)
                y[e] += w * xh[s][lane + 32 * e];
        }
        #pragma unroll
        for (int e = 0; e < 8; ++e) {
            const int idx = lane + 32 * e;
            p2l[t][idx] = learn * relu6f(learn2 * xt[e] + y[e]);
            xh[t][idx]  = xt[e];
        }
    }

    // Final readout with x_query.
    float xv[8], yq[8];
    #pragma unroll
    for (int e = 0; e < 8; ++e) {
        const int idx = lane + 32 * e;
        xv[e] = xq[(size_t)b * MM + idx];
        yq[e] = lampow[TT] * Yb[16 * MM + idx];
    }
    for (int s = 0; s < TT; ++s) {
        float partial = 0.0f;
        #pragma unroll
        for (int e = 0; e < 8; ++e)
            partial += p2l[s][lane + 32 * e] * xv[e];
        const float d = wave_reduce_sum(partial);
        const float w = d * lampow[TT - 1 - s];
        #pragma unroll
        for (int e = 0; e < 8; ++e)
            yq[e] += w * xh[s][lane + 32 * e];
    }
    #pragma unroll
    for (int e = 0; e < 8; ++e) {
        const int idx = lane + 32 * e;
        out[(size_t)b * MM + idx] = relu6f(yq[e]);
    }
}

extern "C" void kernel_launch(void* const* d_in, const int* in_sizes, int n_in,
                              void* d_out, int out_size, void* d_ws, size_t ws_size,
                              hipStream_t stream) {
    const float* A0     = (const float*)d_in[0];  // (B, M, M) initial state
    const float* xs     = (const float*)d_in[1];  // (T, B, M)
    const float* xq     = (const float*)d_in[2];  // (B, M)
    const float* decay  = (const float*)d_in[3];  // (1,)
    const float* learn  = (const float*)d_in[4];  // (1,)
    const float* learn2 = (const float*)d_in[5];  // (1,)
    float* out = (float*)d_out;                   // (B, M)
    float* Y0  = (float*)d_ws;                    // B * 17 * M floats (~4.25 MB)

    memory_phase1_gemm<<<BB, 512, 0, stream>>>(A0, xs, xq, Y0);
    memory_phase2_recur<<<BB, 32, 0, stream>>>(xs, xq, decay, learn, learn2, Y0, out);
}